// MOL_74921409511428
// MI455X (gfx1250) — compile-verified
//
#include <hip/hip_runtime.h>

// Problem sizes fixed by the reference
#define NY 16
#define NP 32
#define RHO 32
#define BS 8192
#define INV_DT 2.0f      // 1/DT, DT = 0.5
#define NEWTON_ITERS 4   // reference: while (i <= MAX_ITER=3) -> at most 4 bodies
#define JACOBI_ITERS 3   // ||D^-1 * 0.05 W^T|| ~ 1e-2 => 3 sweeps ~ 1e-6 rel. error

typedef float v2f __attribute__((ext_vector_type(2)));
typedef float v8f __attribute__((ext_vector_type(8)));

// Per-wave LDS staging tile: 16 rows x 18 floats (pad stride keeps DS banks spread)
#define ROW_STRIDE 18

static __device__ __forceinline__ float fast_rcp(float x) {
  return __builtin_amdgcn_rcpf(x);
}

static __device__ __forceinline__ float fast_tanh(float x) {
  // tanh(x) = 1 - 2/(exp(2x)+1); v_exp_f32 + v_rcp_f32 are TRANS ops that
  // co-execute with the WMMA (XDL) pipe on CDNA5. Saturates correctly at +/-1.
  return 1.0f - 2.0f * fast_rcp(__expf(2.0f * x) + 1.0f);
}

static __device__ __forceinline__ float softplus(float x) {
  return fmaxf(x, 0.0f) + __logf(1.0f + __expf(-fabsf(x)));
}

// Y = X * W on a 16x16 tile via 4 chained V_WMMA_F32_16X16X4_F32 (K = 4*4).
// X arrives in C/D-layout registers (lane (col,hlf) owns X[r+8*hlf][col], r=0..7)
// and is restaged through LDS into the A-operand layout (A: lane m=l%16 holds
// row m; per K-chunk VGPR0/1 = K {0,1} for lanes 0-15 and {2,3} for lanes 16-31).
// bW holds W pre-swizzled into the B-operand layout (loop invariant).
static __device__ __forceinline__ v8f gemm_xw(const float xr[8], const v2f bW[4],
                                              float* lt, int col, int hlf) {
  #pragma unroll
  for (int r = 0; r < 8; ++r)
    lt[(r + 8 * hlf) * ROW_STRIDE + col] = xr[r];
  // Same-wave DS ops retire in order; explicit split-counter wait makes the
  // A-operand reads observe the stores and fences compiler reordering.
  asm volatile("s_wait_dscnt 0" ::: "memory");
  v2f a[4];
  #pragma unroll
  for (int c = 0; c < 4; ++c) {
    const int j0 = 4 * c + 2 * hlf;
    a[c].x = lt[col * ROW_STRIDE + j0];
    a[c].y = lt[col * ROW_STRIDE + j0 + 1];
  }
  asm volatile("s_wait_dscnt 0" ::: "memory");
  v8f acc = {};
  #pragma unroll
  for (int c = 0; c < 4; ++c)
    acc = __builtin_amdgcn_wmma_f32_16x16x4_f32(false, a[c], false, bW[c],
                                                (short)0, acc, false, false);
  return acc;
}

__global__ __launch_bounds__(256) void mol_newton_kernel(
    const float* __restrict__ pDyn,  // (RHO, BS, NP)
    const float* __restrict__ x0,    // (BS, NY)
    const float* __restrict__ W,     // (NY, NY)
    float* __restrict__ ys)          // (RHO, BS, NY)
{
  __shared__ float lds[8][16 * ROW_STRIDE];
  const int lane = threadIdx.x & 31;
  const int wave = threadIdx.x >> 5;
  const int col  = lane & 15;  // N index: state component / B column
  const int hlf  = lane >> 4;  // half-wave select
  const int b0   = (blockIdx.x * 8 + wave) * 16;  // first batch row of tile
  float* lt = lds[wave];

  // W pre-swizzled into the B-operand layout (chunk c: VGPR0 = rows K=4c+{0|2},
  // VGPR1 = rows K=4c+{1|3} selected by half-wave), loaded once, lives in VGPRs.
  v2f bW[4];
  #pragma unroll
  for (int c = 0; c < 4; ++c) {
    const int j0 = 4 * c + 2 * hlf;
    bW[c].x = W[j0 * NY + col];
    bW[c].y = W[(j0 + 1) * NY + col];
  }

  // Carried state (becomes xt each step): lane owns batch b0+r+8*hlf, comp col.
  float x[8];
  #pragma unroll
  for (int r = 0; r < 8; ++r)
    x[r] = x0[(b0 + r + 8 * hlf) * NY + col];

  #pragma unroll 1
  for (int t = 0; t < RHO; ++t) {
    const float* pt = pDyn + (size_t)t * BS * NP;
    if (t + 1 < RHO) {  // pull next step's parameter slab toward L2 (2KB/tile)
      const char* nx =
          (const char*)(pDyn + (size_t)(t + 1) * BS * NP + (size_t)b0 * NP);
      __builtin_prefetch(nx + lane * 64, 0, 1);
    }

    float k[8], src[8], xt[8];
    #pragma unroll
    for (int r = 0; r < 8; ++r) {
      const int b = b0 + r + 8 * hlf;
      k[r]   = softplus(pt[b * NP + col]);
      src[r] = pt[b * NP + NY + col];
      xt[r]  = x[r];
    }

    // Residual g = (x-xt)/DT - src + k*tanh(x) + 0.05*(x@W);
    // d = diagonal of dG/dx = 1/DT + k*sech^2(x).
    float g[8], d[8];
    {
      const v8f y = gemm_xw(x, bW, lt, col, hlf);
      #pragma unroll
      for (int r = 0; r < 8; ++r) {
        const float th = fast_tanh(x[r]);
        g[r] = (x[r] - xt[r]) * INV_DT - src[r] + k[r] * th + 0.05f * y[r];
        d[r] = INV_DT + k[r] * (1.0f - th * th);
      }
    }

    #pragma unroll
    for (int it = 0; it < NEWTON_ITERS; ++it) {
      // Solve (diag(d) + 0.05*W^T) dx = g by Jacobi sweeps: d >= 2 while the
      // off-diagonal entries are ~N(0, 5e-3), so convergence is geometric at
      // rate ~1e-2 — 3 sweeps land far below GTOL. Always-fresh Jacobian
      // diagonal only sharpens the reference's conditional-refresh Newton.
      float dx[8], invd[8];
      #pragma unroll
      for (int r = 0; r < 8; ++r) {
        invd[r] = fast_rcp(d[r]);
        dx[r]   = g[r] * invd[r];
      }
      #pragma unroll
      for (int m = 0; m < JACOBI_ITERS; ++m) {
        const v8f yd = gemm_xw(dx, bW, lt, col, hlf);
        #pragma unroll
        for (int r = 0; r < 8; ++r)
          dx[r] = (g[r] - 0.05f * yd[r]) * invd[r];
      }
      #pragma unroll
      for (int r = 0; r < 8; ++r)
        x[r] -= dx[r];

      if (it + 1 < NEWTON_ITERS) {  // refresh residual + Jacobian diagonal
        const v8f y = gemm_xw(x, bW, lt, col, hlf);
        #pragma unroll
        for (int r = 0; r < 8; ++r) {
          const float th = fast_tanh(x[r]);
          g[r] = (x[r] - xt[r]) * INV_DT - src[r] + k[r] * th + 0.05f * y[r];
          d[r] = INV_DT + k[r] * (1.0f - th * th);
        }
      }
    }

    #pragma unroll
    for (int r = 0; r < 8; ++r)
      ys[((size_t)t * BS + b0 + r + 8 * hlf) * NY + col] = x[r];
  }
}

extern "C" void kernel_launch(void* const* d_in, const int* in_sizes, int n_in,
                              void* d_out, int out_size, void* d_ws, size_t ws_size,
                              hipStream_t stream) {
  (void)in_sizes; (void)n_in; (void)d_ws; (void)ws_size; (void)out_size;
  const float* pDyn = (const float*)d_in[0];  // (RHO, BS, NP) f32
  const float* x0   = (const float*)d_in[1];  // (BS, NY) f32
  const float* W    = (const float*)d_in[2];  // (NY, NY) f32
  float* ys = (float*)d_out;                  // (RHO, BS, NY) f32

  dim3 grid(BS / 128);   // 8 waves/block, 16-batch tile per wave
  dim3 block(256);
  hipLaunchKernelGGL(mol_newton_kernel, grid, block, 0, stream, pDyn, x0, W, ys);
}